// GraphAttentionLayer_47347719471426
// MI455X (gfx1250) — compile-verified
//
#include <hip/hip_runtime.h>

typedef __attribute__((ext_vector_type(2))) float v2f;
typedef __attribute__((ext_vector_type(8))) float v8f;

#define B_    256
#define Fdim  200
#define E1    64
#define E2    128
#define H_    4
#define FP    208     // F padded to multiple of 16
#define HS    132     // LDS row stride (floats) for h-block, avoids 64-bank conflicts
#define ALPHA 0.2f
#define NEGV  -9000000000000000.0f
#define FLTMAX 3.402823466e38f

// One workgroup fuses the full GAT pipeline for one (head, batch) pair.
// h-block (200x128 f32) never leaves LDS; attn rows never leave LDS.
__global__ __launch_bounds__(256, 1)
void gat_fused_kernel(const float* __restrict__ x,
                      const float* __restrict__ adj,
                      const float* __restrict__ W,
                      const float* __restrict__ a,
                      float* __restrict__ out)
{
    __shared__ float sH[FP * HS];                  // 109824 B : h-block (208x128, stride 132)
    __shared__ __align__(16) float sScr[E1 * E2];  //  32768 B : phase1 = W[head]; phase2/3 = si/sj/attn
    float* sW    = sScr;
    float* sSi   = sScr;            // aliases sW after phase 1 (barrier-ordered)
    float* sSj   = sScr + FP;
    float* sAttn = sScr + 2 * FP;   // 16 x FP floats = 13312 B, fits in the 32 KB region

    const int tid  = threadIdx.x;
    const int lane = tid & 31;
    const int wave = tid >> 5;       // 8 waves
    const int half = lane >> 4;      // which 16-lane half (selects K-pair in f32 WMMA layout)
    const int l16  = lane & 15;

    const int hb   = blockIdx.x;     // 0..1023
    const int head = hb >> 8;
    const int b    = hb & 255;

    // ---------------- Phase 0: stage W[head] (64x128) into LDS ----------------
    {
        const float4* w4 = (const float4*)(W + head * E1 * E2);
        float4* s4 = (float4*)sW;
        for (int i = tid; i < (E1 * E2) / 4; i += 256) s4[i] = w4[i];
    }
    __syncthreads();

    // ------- Phase 1: sH = x[b] (200x64) @ W[head] (64x128) via f32 WMMA -------
    // 13 M-tiles of 16 rows; wave w owns M-tiles {w, w+8}. K=64 -> 16 chained k4 steps.
    for (int mt = wave; mt < 13; mt += 8) {
        const int row = mt * 16 + l16;          // A-matrix: lanes 0-15 and 16-31 both map M=l%16
        v2f areg[16];
        if (row < Fdim) {
            const float* xr = x + (b * Fdim + row) * E1;
            #pragma unroll
            for (int t = 0; t < 16; ++t) {
                const int k0 = t * 4 + half * 2; // lane-half picks K-pair {0,1} vs {2,3}
                areg[t].x = xr[k0];
                areg[t].y = xr[k0 + 1];
            }
        } else {
            #pragma unroll
            for (int t = 0; t < 16; ++t) { areg[t].x = 0.f; areg[t].y = 0.f; }
        }
        for (int n = 0; n < 8; ++n) {           // 8 N-tiles cover E2=128
            v8f acc = {};
            #pragma unroll
            for (int t = 0; t < 16; ++t) {
                const int k0 = t * 4 + half * 2;
                v2f breg;                        // B 4x16: row-striped across lanes, half picks K-pair
                breg.x = sW[(k0 + 0) * E2 + n * 16 + l16];
                breg.y = sW[(k0 + 1) * E2 + n * 16 + l16];
                acc = __builtin_amdgcn_wmma_f32_16x16x4_f32(
                          false, areg[t], false, breg, (short)0, acc, false, false);
            }
            #pragma unroll
            for (int v = 0; v < 8; ++v) {        // D: VGPR v -> M = v + half*8
                const int r = mt * 16 + v + half * 8;
                sH[r * HS + n * 16 + l16] = acc[v];
            }
        }
    }
    __syncthreads();

    // ------------- Phase 2: si[f] = h[f,:]·ai, sj[f] = h[f,:]·aj -------------
    {
        const float* ai = a + head * (2 * E2);
        const float* aj = ai + E2;
        for (int r = wave; r < FP; r += 8) {
            float pi = 0.f, pj = 0.f;
            #pragma unroll
            for (int j = 0; j < 4; ++j) {
                const int o = lane + 32 * j;
                const float hv = sH[r * HS + o];
                pi = fmaf(hv, ai[o], pi);
                pj = fmaf(hv, aj[o], pj);
            }
            for (int m = 16; m > 0; m >>= 1) {   // wave32 tree reduce
                pi += __shfl_xor(pi, m, 32);
                pj += __shfl_xor(pj, m, 32);
            }
            if (lane == 0) { sSi[r] = pi; sSj[r] = pj; }
        }
    }
    __syncthreads();

    // --------- Phase 3: per 16-row block: softmax rows, then attn @ h ---------
    for (int rb = 0; rb < 13; ++rb) {
        const int R0 = rb * 16;

        // 3a: each wave builds 2 attention rows (w and w+8) into sAttn
        #pragma unroll
        for (int rsel = 0; rsel < 2; ++rsel) {
            const int rr = wave + rsel * 8;
            const int f  = R0 + rr;
            if (f < Fdim) {
                const float si = sSi[f];
                float tv[7];
                float rmax = -FLTMAX;
                #pragma unroll
                for (int j = 0; j < 7; ++j) {
                    const int c = lane + 32 * j;
                    if (c < Fdim) {
                        const float z  = si + sSj[c];
                        const float e  = (z > 0.f) ? z : ALPHA * z;
                        const float m  = adj[f * Fdim + c];   // L2-resident, shared by all blocks
                        const float mv = (m > 0.f) ? m : NEGV;
                        tv[j] = e * mv;
                        rmax  = fmaxf(rmax, tv[j]);
                    } else {
                        tv[j] = -FLTMAX;
                    }
                }
                for (int m = 16; m > 0; m >>= 1) rmax = fmaxf(rmax, __shfl_xor(rmax, m, 32));
                float pv[7];
                float rsum = 0.f;
                #pragma unroll
                for (int j = 0; j < 7; ++j) {
                    const int c = lane + 32 * j;
                    const float p = (c < Fdim) ? __expf(tv[j] - rmax) : 0.f;
                    pv[j] = p;
                    rsum += p;
                }
                for (int m = 16; m > 0; m >>= 1) rsum += __shfl_xor(rsum, m, 32);
                const float inv = 1.f / rsum;
                #pragma unroll
                for (int j = 0; j < 7; ++j) {
                    const int c = lane + 32 * j;
                    if (c < Fdim) sAttn[rr * FP + c] = pv[j] * inv;
                }
            }
        }
        __syncthreads();

        // 3b: out_tile(16x128) = sAttn(16x200) @ sH(200x128); wave w -> N-tile w
        v8f acc = {};
        for (int kt = 0; kt < 50; ++kt) {        // K=200 = 50 k4 steps, exact
            const int k0 = kt * 4 + half * 2;
            v2f areg, breg;
            areg.x = sAttn[l16 * FP + k0];
            areg.y = sAttn[l16 * FP + k0 + 1];
            breg.x = sH[(k0 + 0) * HS + wave * 16 + l16];
            breg.y = sH[(k0 + 1) * HS + wave * 16 + l16];
            acc = __builtin_amdgcn_wmma_f32_16x16x4_f32(
                      false, areg, false, breg, (short)0, acc, false, false);
        }
        #pragma unroll
        for (int v = 0; v < 8; ++v) {
            const int f = R0 + v + half * 8;
            if (f < Fdim)   // out[b][f][head*128 + col] ; (B,F,H*E2) layout
                out[(b * Fdim + f) * (H_ * E2) + head * E2 + wave * 16 + l16] = acc[v];
        }
        __syncthreads();  // sAttn reused next row-block
    }
}

extern "C" void kernel_launch(void* const* d_in, const int* in_sizes, int n_in,
                              void* d_out, int out_size, void* d_ws, size_t ws_size,
                              hipStream_t stream) {
    const float* x   = (const float*)d_in[0];   // (256,200,64)
    const float* adj = (const float*)d_in[1];   // (200,200)
    const float* W   = (const float*)d_in[2];   // (4,64,128)
    const float* a   = (const float*)d_in[3];   // (4,256)
    float* out = (float*)d_out;                 // (256,200,512)

    dim3 grid(H_ * B_);   // 1024 blocks: one per (head, batch)
    dim3 block(256);      // 8 wave32s
    hipLaunchKernelGGL(gat_fused_kernel, grid, block, 0, stream, x, adj, W, a, out);
}